// MinibatchDiscrimination_50345606644410
// MI455X (gfx1250) — compile-verified
//
#include <hip/hip_runtime.h>
#include <hip/hip_bf16.h>

typedef __attribute__((ext_vector_type(2))) float v2f;
typedef __attribute__((ext_vector_type(8))) float v8f;

#define N_BATCH     256
#define IN_FEAT     2048
#define OUT_FEAT    128
#define KDIMS       32
#define NCOLS       (OUT_FEAT * KDIMS)      // 4096
#define OUT_STRIDE  (IN_FEAT + OUT_FEAT)    // 2176

// ---------------------------------------------------------------------------
// Kernel 1: M[256 x 4096] = x[256 x 2048] @ T[2048 x 4096], fp32 WMMA.
// Each wave computes a 16(M) x 64(N) strip: 4 accumulator tiles, A fragment
// reused across the 4 WMMAs per K-step. K loop: 2048 / 4 = 512 steps.
// Grid: 1024 waves = 128 blocks x 256 threads (8 waves/block).
// ---------------------------------------------------------------------------
__global__ void __launch_bounds__(256)
gemm_f32_wmma(const float* __restrict__ x, const float* __restrict__ T,
              float* __restrict__ M) {
    const int lane = threadIdx.x & 31;
    const int wave = blockIdx.x * 8 + (threadIdx.x >> 5);   // 0..1023
    const int jt   = wave >> 6;                             // 0..15  (M tile)
    const int ng   = wave & 63;                             // 0..63  (64-wide N group)

    const int halfsel = lane >> 4;          // 0: lanes 0-15, 1: lanes 16-31
    const int sub     = lane & 15;

    // A fragment: lane holds A[row][k+koff], A[row][k+koff+1]  (contiguous b64)
    const int arow = jt * 16 + sub;
    const int koff = halfsel * 2;
    const float* __restrict__ xrow = x + (size_t)arow * IN_FEAT;

    // B fragment: lane holds B[k+koff][n], B[k+koff+1][n]
    const int nbase = ng * 64 + sub;

    v8f acc0 = {}, acc1 = {}, acc2 = {}, acc3 = {};

    for (int k = 0; k < IN_FEAT; k += 4) {
        const v2f a = *(const v2f*)(xrow + k + koff);

        const float* __restrict__ b0p = T + (size_t)(k + koff) * NCOLS + nbase;
        const float* __restrict__ b1p = b0p + NCOLS;

        v2f b;
        b.x = b0p[0];  b.y = b1p[0];
        acc0 = __builtin_amdgcn_wmma_f32_16x16x4_f32(false, a, false, b,
                                                     (short)0, acc0, false, false);
        b.x = b0p[16]; b.y = b1p[16];
        acc1 = __builtin_amdgcn_wmma_f32_16x16x4_f32(false, a, false, b,
                                                     (short)0, acc1, false, false);
        b.x = b0p[32]; b.y = b1p[32];
        acc2 = __builtin_amdgcn_wmma_f32_16x16x4_f32(false, a, false, b,
                                                     (short)0, acc2, false, false);
        b.x = b0p[48]; b.y = b1p[48];
        acc3 = __builtin_amdgcn_wmma_f32_16x16x4_f32(false, a, false, b,
                                                     (short)0, acc3, false, false);
    }

    // C/D layout: VGPR v, lanes 0-15 -> M=v, lanes 16-31 -> M=v+8; N = lane&15
    const int mrow0 = jt * 16 + halfsel * 8;
    float* __restrict__ mout = M + (size_t)mrow0 * NCOLS + ng * 64 + sub;
#pragma unroll
    for (int v = 0; v < 8; ++v) {
        float* p = mout + (size_t)v * NCOLS;
        p[0]  = acc0[v];
        p[16] = acc1[v];
        p[32] = acc2[v];
        p[48] = acc3[v];
    }
}

// ---------------------------------------------------------------------------
// Kernel 2: o_b[j,f] = sum_i exp(-sum_k |M[i,f,k]-M[j,f,k]|)
// One block per f (128 blocks, 256 threads). All 256 rows (32 floats each,
// 32 KB) staged in LDS; own row kept in VGPRs; i-loop reads are lane-uniform
// LDS broadcasts.
// ---------------------------------------------------------------------------
__global__ void __launch_bounds__(256)
pairwise_l1_exp(const float* __restrict__ M, float* __restrict__ out) {
    __shared__ float rows[N_BATCH * KDIMS];   // 32 KB

    const int f = blockIdx.x;
    const int j = threadIdx.x;

    const float* __restrict__ mp = M + (size_t)j * NCOLS + f * KDIMS;

    float4 r[8];
#pragma unroll
    for (int kk = 0; kk < 8; ++kk) {
        r[kk] = ((const float4*)mp)[kk];
        ((float4*)(rows + j * KDIMS))[kk] = r[kk];
    }
    __syncthreads();

    float sum = 0.0f;
    for (int i = 0; i < N_BATCH; ++i) {
        const float4* __restrict__ orow = (const float4*)(rows + i * KDIMS);
        float norm = 0.0f;
#pragma unroll
        for (int kk = 0; kk < 8; ++kk) {
            const float4 o = orow[kk];
            norm += __builtin_fabsf(r[kk].x - o.x);
            norm += __builtin_fabsf(r[kk].y - o.y);
            norm += __builtin_fabsf(r[kk].z - o.z);
            norm += __builtin_fabsf(r[kk].w - o.w);
        }
        sum += __expf(-norm);
    }

    out[(size_t)j * OUT_STRIDE + IN_FEAT + f] = sum;
}

// ---------------------------------------------------------------------------
// Kernel 3: out[j, 0:2048] = x[j, :]   (float4 copies, 2176-float row stride)
// 256*512 float4 elements -> 512 blocks x 256 threads.
// ---------------------------------------------------------------------------
__global__ void __launch_bounds__(256)
copy_x_rows(const float* __restrict__ x, float* __restrict__ out) {
    const int idx = blockIdx.x * blockDim.x + threadIdx.x;  // 0 .. 256*512-1
    const int j = idx >> 9;       // / 512
    const int c = idx & 511;      // float4 column within first 2048 floats
    const float4 v = ((const float4*)(x + (size_t)j * IN_FEAT))[c];
    ((float4*)(out + (size_t)j * OUT_STRIDE))[c] = v;
}

// ---------------------------------------------------------------------------
extern "C" void kernel_launch(void* const* d_in, const int* in_sizes, int n_in,
                              void* d_out, int out_size, void* d_ws, size_t ws_size,
                              hipStream_t stream) {
    (void)in_sizes; (void)n_in; (void)out_size; (void)ws_size;

    const float* x = (const float*)d_in[0];   // [256, 2048] f32
    const float* T = (const float*)d_in[1];   // [2048, 128, 32] f32
    float* out = (float*)d_out;               // [256, 2176] f32
    float* M   = (float*)d_ws;                // [256, 4096] f32 scratch (4 MB)

    // GEMM: 1024 waves -> 128 blocks of 256 threads
    gemm_f32_wmma<<<128, 256, 0, stream>>>(x, T, M);

    // Pairwise exp(-L1): one block per output feature
    pairwise_l1_exp<<<OUT_FEAT, 256, 0, stream>>>(M, out);

    // Concat x into the output (independent of the other kernels)
    copy_x_rows<<<512, 256, 0, stream>>>(x, out);
}